// FaceEncoder2_76416058130791
// MI455X (gfx1250) — compile-verified
//
#include <hip/hip_runtime.h>
#include <math.h>

#define B_    128
#define N_    468
#define BN    (B_*N_)            // 59904
#define EPER  2800
#define ETOT  (B_*EPER + BN)     // 418304
#define HEADS 4
#define HID   128
#define HC    (HEADS*HID)        // 512
#define OUTD  128

typedef __attribute__((ext_vector_type(2))) float v2f;
typedef __attribute__((ext_vector_type(4))) float v4f;
typedef __attribute__((ext_vector_type(8))) float v8f;
typedef __attribute__((ext_vector_type(4))) int   v4i;

// pointer-to-int4 in explicit address spaces (match builtin signature:
// arg0 = int4 __device__* (AS1 global), arg1 = int4 AS3 (LDS))
typedef __attribute__((address_space(1))) v4i* gv4i_p;
typedef __attribute__((address_space(3))) v4i* sv4i_p;

// ---- CDNA5 async global->LDS copy (ASYNCcnt-tracked), feature-guarded ----
#if defined(__has_builtin)
#  if __has_builtin(__builtin_amdgcn_global_load_async_to_lds_b128)
#    define HAVE_ASYNC_LDS 1
#  endif
#endif

__device__ __forceinline__ void wait_asynccnt0() {
#if defined(__has_builtin) && __has_builtin(__builtin_amdgcn_s_wait_asynccnt)
    __builtin_amdgcn_s_wait_asynccnt(0);
#else
    asm volatile("s_wait_asynccnt 0" ::: "memory");
#endif
}

// ---------- helpers ----------
__device__ __forceinline__ float lrelu(float v) { return v > 0.0f ? v : 0.2f * v; }

// order-preserving float -> uint key (for atomicMax-based segment max)
__device__ __forceinline__ unsigned fkey(float f) {
    unsigned u = __float_as_uint(f);
    return (u & 0x80000000u) ? ~u : (u | 0x80000000u);
}
__device__ __forceinline__ float funkey(unsigned u) {
    return __uint_as_float((u & 0x80000000u) ? (u ^ 0x80000000u) : ~u);
}

// ---------- generic zero ----------
__global__ void zero_kernel(float* __restrict__ p, size_t n) {
    size_t i = (size_t)blockIdx.x * blockDim.x + threadIdx.x;
    size_t stride = (size_t)gridDim.x * blockDim.x;
    for (; i < n; i += stride) p[i] = 0.0f;
}

// ---------- 1) per-batch normalize (mean / unbiased std over N) ----------
__global__ void normalize_kernel(const float* __restrict__ lm, float* __restrict__ x) {
    __shared__ float s0[256], s1[256], q0[256], q1[256];
    __shared__ float mean0, mean1, inv0, inv1;
    const int b = blockIdx.x, t = threadIdx.x;
    float a0 = 0.f, a1 = 0.f, c0 = 0.f, c1 = 0.f;
    for (int i = t; i < N_; i += 256) {
        float v0 = lm[(b * N_ + i) * 2 + 0];
        float v1 = lm[(b * N_ + i) * 2 + 1];
        a0 += v0; a1 += v1; c0 += v0 * v0; c1 += v1 * v1;
    }
    s0[t] = a0; s1[t] = a1; q0[t] = c0; q1[t] = c1;
    __syncthreads();
    for (int o = 128; o > 0; o >>= 1) {
        if (t < o) { s0[t] += s0[t + o]; s1[t] += s1[t + o];
                     q0[t] += q0[t + o]; q1[t] += q1[t + o]; }
        __syncthreads();
    }
    if (t == 0) {
        float m0 = s0[0] / N_, m1 = s1[0] / N_;
        float var0 = (q0[0] - N_ * m0 * m0) / (N_ - 1);
        float var1 = (q1[0] - N_ * m1 * m1) / (N_ - 1);
        mean0 = m0; mean1 = m1;
        inv0 = 1.0f / sqrtf(var0); inv1 = 1.0f / sqrtf(var1);
    }
    __syncthreads();
    for (int i = t; i < N_; i += 256) {
        x[(b * N_ + i) * 2 + 0] = (lm[(b * N_ + i) * 2 + 0] - mean0) * inv0;
        x[(b * N_ + i) * 2 + 1] = (lm[(b * N_ + i) * 2 + 1] - mean1) * inv1;
    }
}

// ---------- 2) layer-1 projection (K=2, pure VALU) ----------
__global__ void gemm1_kernel(const float* __restrict__ x, const float* __restrict__ W1,
                             float* __restrict__ H) {
    int t = blockIdx.x * blockDim.x + threadIdx.x;
    if (t >= BN * HC) return;
    int n = t / HC, j = t - n * HC;
    H[t] = fmaf(x[n * 2 + 0], W1[j], x[n * 2 + 1] * W1[HC + j]);
}

// ---------- 3) per-node attention logits: alpha = <h[n,h,:], a[h,:]> ----------
__global__ void alpha_kernel(const float* __restrict__ H,
                             const float* __restrict__ a_src, const float* __restrict__ a_dst,
                             float* __restrict__ AS, float* __restrict__ AD) {
    int t = blockIdx.x * blockDim.x + threadIdx.x;
    if (t >= BN * HEADS) return;
    int n = t / HEADS, h = t - n * HEADS;
    const float* hp = H + (size_t)n * HC + h * HID;
    const float* as = a_src + h * HID;
    const float* ad = a_dst + h * HID;
    float vs = 0.f, vd = 0.f;
    for (int c = 0; c < HID; ++c) { float hv = hp[c]; vs = fmaf(hv, as[c], vs); vd = fmaf(hv, ad[c], vd); }
    AS[t] = vs; AD[t] = vd;
}

// ---------- 4) edge pass A: segment max over dst ----------
__global__ void edge_max_kernel(const int* __restrict__ src, const int* __restrict__ dst,
                                const float* __restrict__ AS, const float* __restrict__ AD,
                                unsigned* __restrict__ EM) {
    int e = blockIdx.x * blockDim.x + threadIdx.x;
    if (e >= ETOT) return;
    int s = src[e], d = dst[e];
#pragma unroll
    for (int h = 0; h < HEADS; ++h) {
        float v = lrelu(AS[s * HEADS + h] + AD[d * HEADS + h]);
        atomicMax(&EM[d * HEADS + h], fkey(v));
    }
}

// ---------- 5) edge pass B: exp-sum over dst ----------
__global__ void edge_expsum_kernel(const int* __restrict__ src, const int* __restrict__ dst,
                                   const float* __restrict__ AS, const float* __restrict__ AD,
                                   const unsigned* __restrict__ EM, float* __restrict__ DEN) {
    int e = blockIdx.x * blockDim.x + threadIdx.x;
    if (e >= ETOT) return;
    int s = src[e], d = dst[e];
#pragma unroll
    for (int h = 0; h < HEADS; ++h) {
        float v = lrelu(AS[s * HEADS + h] + AD[d * HEADS + h]);
        float m = funkey(EM[d * HEADS + h]);
        atomicAdd(&DEN[d * HEADS + h], __expf(v - m));
    }
}

// ---------- 6) edge pass C: weighted message scatter (one wave32 per edge) ----------
__global__ void edge_agg_kernel(const int* __restrict__ src, const int* __restrict__ dst,
                                const float* __restrict__ AS, const float* __restrict__ AD,
                                const unsigned* __restrict__ EM, const float* __restrict__ DEN,
                                const float* __restrict__ H, float* __restrict__ AGG) {
    int wid = (blockIdx.x * blockDim.x + threadIdx.x) >> 5;
    int lane = threadIdx.x & 31;
    if (wid >= ETOT) return;
    int s = src[wid], d = dst[wid];
    const float* hs = H + (size_t)s * HC;
    float* ag = AGG + (size_t)d * HC;
    __builtin_prefetch(&hs[lane * 16], 0, 3);   // global_prefetch_b8: pull h[src] row
    float coef[HEADS];
#pragma unroll
    for (int h = 0; h < HEADS; ++h) {
        float v = lrelu(AS[s * HEADS + h] + AD[d * HEADS + h]);
        float m = funkey(EM[d * HEADS + h]);
        coef[h] = __expf(v - m) / (DEN[d * HEADS + h] + 1e-16f);
    }
#pragma unroll
    for (int h = 0; h < HEADS; ++h) {
        float c = coef[h];
#pragma unroll
        for (int t = 0; t < HID / 32; ++t) {
            int idx = h * HID + t * 32 + lane;
            atomicAdd(&ag[idx], hs[idx] * c);
        }
    }
}

// ---------- 7) head-mean + bias + relu ----------
__global__ void mean_bias_relu_kernel(const float* __restrict__ AGG, const float* __restrict__ bias,
                                      float* __restrict__ X2) {
    int t = blockIdx.x * blockDim.x + threadIdx.x;
    if (t >= BN * OUTD) return;
    int n = t / OUTD, c = t - n * OUTD;
    const float* a = AGG + (size_t)n * HC;
    float v = 0.25f * (a[c] + a[HID + c] + a[2 * HID + c] + a[3 * HID + c]) + bias[c];
    X2[t] = fmaxf(v, 0.0f);
}

// ---------- 8) layer-2 GEMM: f32 WMMA (16x16x4) with async-LDS A staging ----------
// A = X2 [BN,128] row-major, B = W2 [128,512] row-major, D = H [BN,512].
// One block (8 waves) per 16-row tile. The 16x128 A tile is a CONTIGUOUS 8 KB
// region of X2; it is staged once into LDS via GLOBAL_LOAD_ASYNC_TO_LDS_B128
// (ASYNCcnt) and consumed by all 8 waves across all 32 column tiles (8x less
// A traffic than reloading per column group). LDS row stride padded to 132
// floats so row*132 mod 64 = 4*row -> bank-conflict-free A reads.
// No divergence anywhere: EXEC stays all-ones for WMMA.
#define APAD  132
__global__ void gemm2_wmma_kernel(const float* __restrict__ A, const float* __restrict__ Bm,
                                  float* __restrict__ D) {
    __shared__ float Atile[16 * APAD];          // 8.25 KB
    const int tid  = threadIdx.x;               // 256 threads = 8 waves
    const int n0   = blockIdx.x * 16;
    const float* Ablk = A + (size_t)n0 * HID;   // contiguous 16*128 floats

    // stage A tile: 512 x 16-byte chunks, 2 per thread (row-padded LDS dst)
#pragma unroll
    for (int rep = 0; rep < 2; ++rep) {
        int tr  = tid + rep * 256;              // 0..511
        int row = tr >> 5;                      // 0..15
        int c   = tr & 31;                      // 16B chunk within row
#if defined(HAVE_ASYNC_LDS)
        __builtin_amdgcn_global_load_async_to_lds_b128(
            (gv4i_p)(Ablk + row * HID + c * 4),
            (sv4i_p)(Atile + row * APAD + c * 4),
            0, 0);
#else
        *(v4f*)(Atile + row * APAD + c * 4) = *(const v4f*)(Ablk + row * HID + c * 4);
#endif
    }
#if defined(HAVE_ASYNC_LDS)
    wait_asynccnt0();
#endif
    __syncthreads();

    const int wave = tid >> 5;                  // 8 waves -> 4 col tiles each
    const int lane = tid & 31;
    const int half = lane >> 4;                 // 0: K pair {0,1}; 1: K pair {2,3}
    const int l16  = lane & 15;
    const float* arow = Atile + l16 * APAD;     // A row for this lane (LDS)

#pragma unroll
    for (int i = 0; i < 4; ++i) {
        const int j0   = (wave * 4 + i) * 16;   // column tile
        const int bcol = j0 + l16;
        v8f acc = {};
        for (int k = 0; k < HID; k += 4) {
            const int ka = k + 2 * half;
            v2f a, b;
            a.x = arow[ka];
            a.y = arow[ka + 1];
            b.x = Bm[ka * HC + bcol];
            b.y = Bm[(ka + 1) * HC + bcol];
            acc = __builtin_amdgcn_wmma_f32_16x16x4_f32(
                /*neg_a=*/false, a, /*neg_b=*/false, b,
                /*c_mod=*/(short)0, acc, /*reuse_a=*/false, /*reuse_b=*/false);
        }
#pragma unroll
        for (int r = 0; r < 8; ++r) {
            int row = n0 + r + 8 * half;        // C/D: VGPR r -> M=r (lanes 0-15), M=r+8 (16-31)
            D[(size_t)row * HC + bcol] = acc[r];
        }
    }
}

// ---------- 9) gate logits + per-graph max ----------
__global__ void gate_kernel(const float* __restrict__ X3, const float* __restrict__ gw,
                            const float* __restrict__ gb, float* __restrict__ GATE,
                            unsigned* __restrict__ GMAX) {
    int n = blockIdx.x * blockDim.x + threadIdx.x;
    if (n >= BN) return;
    const float* xp = X3 + (size_t)n * OUTD;
    float g = gb[0];
    for (int c = 0; c < OUTD; ++c) g = fmaf(xp[c], gw[c], g);
    GATE[n] = g;
    atomicMax(&GMAX[n / N_], fkey(g));
}

// ---------- 10) gate exp + per-graph denom ----------
__global__ void gate_exp_kernel(float* __restrict__ GATE, const unsigned* __restrict__ GMAX,
                                float* __restrict__ GDEN) {
    int n = blockIdx.x * blockDim.x + threadIdx.x;
    if (n >= BN) return;
    int b = n / N_;
    float e = __expf(GATE[n] - funkey(GMAX[b]));
    GATE[n] = e;
    atomicAdd(&GDEN[b], e);
}

// ---------- 11) attention-weighted pool (one wave per node) ----------
__global__ void pool_kernel(const float* __restrict__ GATE, const float* __restrict__ GDEN,
                            const float* __restrict__ X3, float* __restrict__ out) {
    int wid = (blockIdx.x * blockDim.x + threadIdx.x) >> 5;
    int lane = threadIdx.x & 31;
    if (wid >= BN) return;
    int b = wid / N_;
    float att = GATE[wid] / (GDEN[b] + 1e-16f);
    const float* xp = X3 + (size_t)wid * OUTD;
    float* op = out + (size_t)b * OUTD;
#pragma unroll
    for (int t = 0; t < OUTD / 32; ++t) {
        int c = t * 32 + lane;
        atomicAdd(&op[c], att * xp[c]);
    }
}

// ---------- launch ----------
static inline void zero_async(float* p, size_t n, hipStream_t s) {
    zero_kernel<<<2048, 256, 0, s>>>(p, n);
}

extern "C" void kernel_launch(void* const* d_in, const int* in_sizes, int n_in,
                              void* d_out, int out_size, void* d_ws, size_t ws_size,
                              hipStream_t stream) {
    const float* landmark = (const float*)d_in[0];
    const int*   eidx     = (const int*)d_in[1];
    const float* W1       = (const float*)d_in[2];
    const float* a_src1   = (const float*)d_in[3];
    const float* a_dst1   = (const float*)d_in[4];
    const float* b1       = (const float*)d_in[5];
    const float* W2       = (const float*)d_in[6];
    const float* a_src2   = (const float*)d_in[7];
    const float* a_dst2   = (const float*)d_in[8];
    const float* b2       = (const float*)d_in[9];
    const float* gate_w   = (const float*)d_in[10];
    const float* gate_b   = (const float*)d_in[11];
    float* out = (float*)d_out;

    const int* esrc = eidx;
    const int* edst = eidx + ETOT;

    // workspace carve
    float* X   = (float*)d_ws;                      // BN*2
    float* H   = X   + (size_t)BN * 2;              // BN*512
    float* AGG = H   + (size_t)BN * HC;             // BN*512
    float* X2  = AGG + (size_t)BN * HC;             // BN*128
    float* X3  = X2  + (size_t)BN * OUTD;           // BN*128
    float* AS  = X3  + (size_t)BN * OUTD;           // BN*4
    float* AD  = AS  + (size_t)BN * HEADS;          // BN*4
    unsigned* EM = (unsigned*)(AD + (size_t)BN * HEADS); // BN*4 (uint keys)
    float* DEN = (float*)(EM + (size_t)BN * HEADS); // BN*4
    float* GATE = DEN + (size_t)BN * HEADS;         // BN
    unsigned* GMAX = (unsigned*)(GATE + BN);        // B
    float* GDEN = (float*)(GMAX + B_);              // B

    const int T = 256;
    const int eb   = (ETOT + T - 1) / T;            // thread-per-edge grids
    const int ewb  = (ETOT * 32 + T - 1) / T;       // wave-per-edge grids
    const int nhb  = (BN * HEADS + T - 1) / T;
    const int ncb  = (BN * OUTD + T - 1) / T;
    const int nb   = (BN + T - 1) / T;
    const int nwb  = (BN * 32 + T - 1) / T;

    // 0) normalize
    normalize_kernel<<<B_, 256, 0, stream>>>(landmark, X);

    // ---- layer 1 ----
    gemm1_kernel<<<(BN * HC + T - 1) / T, T, 0, stream>>>(X, W1, H);
    alpha_kernel<<<nhb, T, 0, stream>>>(H, a_src1, a_dst1, AS, AD);
    zero_async((float*)EM, (size_t)2 * BN * HEADS, stream);   // EM keys + DEN
    zero_async(AGG, (size_t)BN * HC, stream);
    edge_max_kernel<<<eb, T, 0, stream>>>(esrc, edst, AS, AD, EM);
    edge_expsum_kernel<<<eb, T, 0, stream>>>(esrc, edst, AS, AD, EM, DEN);
    edge_agg_kernel<<<ewb, T, 0, stream>>>(esrc, edst, AS, AD, EM, DEN, H, AGG);
    mean_bias_relu_kernel<<<ncb, T, 0, stream>>>(AGG, b1, X2);

    // ---- layer 2 ----
    gemm2_wmma_kernel<<<BN / 16, 256, 0, stream>>>(X2, W2, H);
    alpha_kernel<<<nhb, T, 0, stream>>>(H, a_src2, a_dst2, AS, AD);
    zero_async((float*)EM, (size_t)2 * BN * HEADS, stream);
    zero_async(AGG, (size_t)BN * HC, stream);
    edge_max_kernel<<<eb, T, 0, stream>>>(esrc, edst, AS, AD, EM);
    edge_expsum_kernel<<<eb, T, 0, stream>>>(esrc, edst, AS, AD, EM, DEN);
    edge_agg_kernel<<<ewb, T, 0, stream>>>(esrc, edst, AS, AD, EM, DEN, H, AGG);
    mean_bias_relu_kernel<<<ncb, T, 0, stream>>>(AGG, b2, X3);

    // ---- global attention pool ----
    zero_async((float*)GMAX, (size_t)2 * B_, stream);         // GMAX keys + GDEN
    zero_async(out, (size_t)B_ * OUTD, stream);
    gate_kernel<<<nb, T, 0, stream>>>(X3, gate_w, gate_b, GATE, GMAX);
    gate_exp_kernel<<<nb, T, 0, stream>>>(GATE, GMAX, GDEN);
    pool_kernel<<<nwb, T, 0, stream>>>(GATE, GDEN, X3, out);
}